// GCNConvScatterGather_4629974745746
// MI455X (gfx1250) — compile-verified
//
#include <hip/hip_runtime.h>

typedef float v2f __attribute__((ext_vector_type(2)));
typedef float v8f __attribute__((ext_vector_type(8)));

#define N_NODES 50000
#define IN_CH   128
#define OUT_CH  64
#define N_EDGES 800000

// ---------------------------------------------------------------------------
// Kernel 1: out[n, c] = bias[c]   (fuses "+ bias" into the accumulator init)
// ---------------------------------------------------------------------------
__global__ void init_bias_kernel(float* __restrict__ out,
                                 const float* __restrict__ bias) {
    int i = blockIdx.x * blockDim.x + threadIdx.x;
    if (i < N_NODES * OUT_CH) out[i] = bias[i & (OUT_CH - 1)];
}

// ---------------------------------------------------------------------------
// Kernel 2: h = x @ W^T  via V_WMMA_F32_16X16X4_F32
//   One wave computes one 16x16 tile of h. 12500 tiles, 4 waves per block.
//   A (16x4 f32): lane (16*half + l) holds float2 x[m0+l][k0+2*half ..]
//   B (4x16 f32): lane (16*half + l) holds float2 W[n0+l][k0+2*half ..]
//   D (16x16 f32): vgpr r of lane (16*half + l) is h[m0+8*half+r][n0+l]
// ---------------------------------------------------------------------------
__global__ __launch_bounds__(128) void gemm_xwT_wmma(
        const float* __restrict__ x,
        const float* __restrict__ W,
        float* __restrict__ h) {
    const int wave = threadIdx.x >> 5;
    const int lane = threadIdx.x & 31;
    const int tile = blockIdx.x * 4 + wave;     // 0 .. 12499
    const int tm   = tile >> 2;                 // 0 .. 3124  (M tile)
    const int tn   = tile & 3;                  // 0 .. 3     (N tile)
    const int l    = lane & 15;
    const int half = lane >> 4;

    const float* arow = x + (tm * 16 + l) * IN_CH + half * 2;
    const float* brow = W + (tn * 16 + l) * IN_CH + half * 2;

    v8f c = {};
#pragma unroll
    for (int k0 = 0; k0 < IN_CH; k0 += 4) {
        v2f a = *(const v2f*)(arow + k0);
        v2f b = *(const v2f*)(brow + k0);
        // (neg_a, A, neg_b, B, c_mod, C, reuse_a, reuse_b)
        c = __builtin_amdgcn_wmma_f32_16x16x4_f32(false, a, false, b,
                                                  (short)0, c, false, false);
    }

    float* drow = h + (tm * 16 + half * 8) * OUT_CH + tn * 16 + l;
#pragma unroll
    for (int r = 0; r < 8; ++r)
        drow[r * OUT_CH] = c[r];
}

// ---------------------------------------------------------------------------
// Kernel 3: scatter-add. 64 consecutive threads own one edge (one channel
// each). The edge id is wave-uniform (a wave covers channels 0-31 or 32-63
// of one edge), so force it scalar with readfirstlane: the two int64 index
// fetches become s_load_b64 (SMEM), and src/dst become SGPRs so the gather
// and the atomic use saddr + per-lane voffset. The atomic is pinned to the
// non-returning global_atomic_add_f32 (STOREcnt path) via inline asm; h and
// out (12.8 MB each) are L2-resident (192 MB L2), so this is one coalesced
// 128B L2 atomic burst per wave.
// ---------------------------------------------------------------------------
__global__ __launch_bounds__(256) void scatter_add_kernel(
        const float* __restrict__ h,
        const long long* __restrict__ ei,   // [2, N_EDGES] int64
        float* __restrict__ out) {
    const int lane      = threadIdx.x & 31;
    const int wave_base = blockIdx.x * 256 + (threadIdx.x & ~31); // 32-aligned
    int e = __builtin_amdgcn_readfirstlane(wave_base >> 6);       // wave-uniform
    if (e >= N_EDGES) return;
    const int c = (wave_base & 63) | lane;                        // channel 0..63

    const int src = (int)ei[e];               // scalar -> s_load_b64
    const int dst = (int)ei[N_EDGES + e];     // scalar -> s_load_b64

    const float val = h[src * OUT_CH + c];    // saddr + voffset gather

    const float* base = out + dst * OUT_CH;   // uniform SGPR-pair base
    unsigned int voff = (unsigned int)(c * 4);
    asm volatile("global_atomic_add_f32 %0, %1, %2"
                 :
                 : "v"(voff), "v"(val), "s"(base)
                 : "memory");
}

// ---------------------------------------------------------------------------
extern "C" void kernel_launch(void* const* d_in, const int* in_sizes, int n_in,
                              void* d_out, int out_size, void* d_ws, size_t ws_size,
                              hipStream_t stream) {
    const float*     x    = (const float*)d_in[0];
    const long long* ei   = (const long long*)d_in[1];
    const float*     W    = (const float*)d_in[2];
    const float*     bias = (const float*)d_in[3];
    float* out = (float*)d_out;
    float* h   = (float*)d_ws;          // 50000*64*4 = 12.8 MB scratch

    // out = bias (broadcast)
    init_bias_kernel<<<(N_NODES * OUT_CH + 255) / 256, 256, 0, stream>>>(out, bias);

    // h = x @ W^T   (12500 16x16 tiles, 4 waves/block)
    gemm_xwT_wmma<<<(N_NODES / 16) * (OUT_CH / 16) / 4, 128, 0, stream>>>(x, W, h);

    // out[dst,:] += h[src,:]  for every edge
    scatter_add_kernel<<<(N_EDGES * OUT_CH) / 256, 256, 0, stream>>>(h, ei, out);
}